// GraphAttentionLayer_31662498906307
// MI455X (gfx1250) — compile-verified
//
#include <hip/hip_runtime.h>
#include <hip/hip_bf16.h>

// Problem constants (match reference)
#define BB     4
#define NN     2048
#define INF_   128
#define HH     4
#define OF     64
#define CC     256   // H*OF

typedef __attribute__((ext_vector_type(2))) float v2f;
typedef __attribute__((ext_vector_type(8))) float v8f;

__device__ __forceinline__ v8f wmma_f32_4(v2f a, v2f b, v8f c) {
  // V_WMMA_F32_16X16X4_F32 : D = A(16x4) x B(4x16) + C(16x16)
  return __builtin_amdgcn_wmma_f32_16x16x4_f32(
      /*neg_a=*/false, a, /*neg_b=*/false, b,
      /*c_mod=*/(short)0, c, /*reuse_a=*/false, /*reuse_b=*/false);
}

// ---------------------------------------------------------------------------
// Kernel 1: Wh = x @ W   (8192x128 @ 128x256), f32 WMMA 16x16x4.
// One wave per 16x16 output tile; 4 waves per block.
// A layout: lane&15 = M row, lanes 0-15 hold K=(4t,4t+1), lanes 16-31 K=(4t+2,4t+3).
// B layout: lane&15 = N col, same K split.
// ---------------------------------------------------------------------------
__global__ void __launch_bounds__(128)
wh_gemm(const float* __restrict__ x, const float* __restrict__ W,
        float* __restrict__ Wh) {
  const int tile = blockIdx.x * 4 + (threadIdx.x >> 5);   // 8192 tiles
  const int mt = tile >> 4;            // 512 row tiles
  const int nt = tile & 15;            // 16 col tiles
  const int row0 = mt * 16;
  const int f0   = nt * 16;
  const int lane = threadIdx.x & 31;
  const int ml   = lane & 15;
  const int half = lane >> 4;

  const float* __restrict__ xrow = x + (size_t)(row0 + ml) * INF_;
  v8f acc = {};
  for (int t = 0; t < INF_ / 4; ++t) {
    const int kb = t * 4 + half * 2;
    v2f a; a.x = xrow[kb]; a.y = xrow[kb + 1];
    v2f b; b.x = W[(size_t)kb * CC + f0 + ml];
           b.y = W[(size_t)(kb + 1) * CC + f0 + ml];
    acc = wmma_f32_4(a, b, acc);
  }
  // C/D layout: vgpr v -> M = v + 8*half ; N = lane&15
#pragma unroll
  for (int v = 0; v < 8; ++v) {
    const int r = row0 + v + half * 8;
    Wh[(size_t)r * CC + f0 + ml] = acc[v];
  }
}

// ---------------------------------------------------------------------------
// Kernel 2: e_dst[row,h] = sum_f Wh[row, h*64+f] * a[(64+f)*H + h]
// 128 threads/block = 4 waves, wave w handles head h=w; wave32 shuffle reduce.
// ---------------------------------------------------------------------------
__global__ void __launch_bounds__(128)
edst_kernel(const float* __restrict__ Wh, const float* __restrict__ a,
            float* __restrict__ ed) {
  const int row  = blockIdx.x;               // 0..8191
  const int h    = threadIdx.x >> 5;
  const int lane = threadIdx.x & 31;
  const float* __restrict__ w = Wh + (size_t)row * CC + h * OF;
  float s = w[lane]      * a[(size_t)(OF + lane) * HH + h]
          + w[lane + 32] * a[(size_t)(OF + lane + 32) * HH + h];
#pragma unroll
  for (int off = 16; off > 0; off >>= 1) s += __shfl_down(s, off, 32);
  if (lane == 0) ed[(size_t)row * HH + h] = s;
}

// ---------------------------------------------------------------------------
// Kernel 3: per (b,i) masked softmax stats over j for all 4 heads:
//   m[h] = max_{j: adj!=0} e_dst[b,j,h] ;  D[h] = sum exp(e_dst - m)
// ---------------------------------------------------------------------------
__global__ void __launch_bounds__(256)
row_stats(const int* __restrict__ adj, const float* __restrict__ ed,
          float* __restrict__ mOut, float* __restrict__ dOut) {
  const int row = blockIdx.x;                  // b*N + i
  const int b   = row >> 11;                   // / 2048
  const int* __restrict__  arow = adj + (size_t)row * NN;
  const float* __restrict__ edb = ed + (size_t)b * NN * HH;
  const int t = threadIdx.x;

  __shared__ float4 red[256];
  const float NEG = -__builtin_inff();

  float4 mx = {NEG, NEG, NEG, NEG};
  for (int j = t; j < NN; j += 256) {
    const int ad = arow[j];
    const float4 e = *(const float4*)(edb + (size_t)j * HH);
    if (ad != 0) {
      mx.x = fmaxf(mx.x, e.x); mx.y = fmaxf(mx.y, e.y);
      mx.z = fmaxf(mx.z, e.z); mx.w = fmaxf(mx.w, e.w);
    }
  }
  red[t] = mx;
  __syncthreads();
  for (int s = 128; s > 0; s >>= 1) {
    if (t < s) {
      float4 o = red[t + s];
      red[t].x = fmaxf(red[t].x, o.x); red[t].y = fmaxf(red[t].y, o.y);
      red[t].z = fmaxf(red[t].z, o.z); red[t].w = fmaxf(red[t].w, o.w);
    }
    __syncthreads();
  }
  const float4 m = red[0];
  __syncthreads();

  float4 sm = {0.f, 0.f, 0.f, 0.f};
  for (int j = t; j < NN; j += 256) {
    const int ad = arow[j];
    const float4 e = *(const float4*)(edb + (size_t)j * HH);
    if (ad != 0) {
      sm.x += __expf(e.x - m.x); sm.y += __expf(e.y - m.y);
      sm.z += __expf(e.z - m.z); sm.w += __expf(e.w - m.w);
    }
  }
  red[t] = sm;
  __syncthreads();
  for (int s = 128; s > 0; s >>= 1) {
    if (t < s) {
      float4 o = red[t + s];
      red[t].x += o.x; red[t].y += o.y; red[t].z += o.z; red[t].w += o.w;
    }
    __syncthreads();
  }
  if (t == 0) {
    *(float4*)(mOut + (size_t)row * HH) = m;
    *(float4*)(dOut + (size_t)row * HH) = red[0];
  }
}

// ---------------------------------------------------------------------------
// Kernel 4 (fused): per (b, 16-row i-tile): build probability tile P[16x16x4h]
// in LDS, write attention coalesced, and accumulate h_prime with f32 WMMA.
// 128 threads = 4 waves; wave w handles head h=w (16x64 output slab).
// ---------------------------------------------------------------------------
__global__ void __launch_bounds__(128)
attn_agg(const int* __restrict__ adj, const float* __restrict__ ed,
         const float* __restrict__ mS, const float* __restrict__ dS,
         const float* __restrict__ Wh,
         float* __restrict__ out, float* __restrict__ attn) {
  const int blk = blockIdx.x;               // 512 = B * N/16
  const int b   = blk >> 7;
  const int i0  = (blk & 127) * 16;
  const int t    = threadIdx.x;
  const int h    = t >> 5;
  const int lane = t & 31;
  const int ml   = lane & 15;
  const int half = lane >> 4;

  __shared__ int   s_adj[256];          // 16x16 adj tile  [m*16 + k]
  __shared__ float s_p[16 * 64];        // P tile          [m*64 + k*4 + h]
  __shared__ float s_m[HH][16], s_d[HH][16];

  if (lane < 16) {
    const int r = b * NN + i0 + lane;
    s_m[h][lane] = mS[(size_t)r * HH + h];
    s_d[h][lane] = dS[(size_t)r * HH + h];
  }
  __syncthreads();

  const float* __restrict__ edb  = ed  + (size_t)b * NN * HH;
  const int*   __restrict__ adjb = adj + (size_t)(b * NN + i0) * NN;
  const float* __restrict__ whb  = Wh  + (size_t)b * NN * CC;

  v8f acc0 = {}, acc1 = {}, acc2 = {}, acc3 = {};

  for (int j0 = 0; j0 < NN; j0 += 16) {
    // cooperative 16x16 adj tile load
#pragma unroll
    for (int r = 0; r < 2; ++r) {
      const int idx = t + r * 128;
      s_adj[idx] = adjb[(size_t)(idx >> 4) * NN + j0 + (idx & 15)];
    }
    __syncthreads();

    // each wave fills its head's plane of P: 8 elements per lane
#pragma unroll
    for (int r = 0; r < 8; ++r) {
      const int idx = lane + r * 32;
      const int mm = idx >> 4, kk = idx & 15;
      const float e = edb[(size_t)(j0 + kk) * HH + h];
      const float p = s_adj[idx]
                    ? __expf(e - s_m[h][mm]) / s_d[h][mm] : 0.0f;
      s_p[mm * 64 + kk * 4 + h] = p;
    }
    __syncthreads();

    // coalesced attention write: row m gets 64 contiguous floats (16 j x 4 h)
    float* __restrict__ arow = attn + ((size_t)(b * NN + i0) * NN + j0) * HH;
#pragma unroll
    for (int r = 0; r < 8; ++r) {
      const int gi = t + r * 128;               // [m*64 + c]
      arow[(size_t)(gi >> 6) * NN * HH + (gi & 63)] = s_p[gi];
    }

    // WMMA accumulation: A = P(16i x 16j), B = Wh(16j x 64f), K-steps of 4
#pragma unroll
    for (int kt = 0; kt < 4; ++kt) {
      const int kb = kt * 4 + half * 2;
      v2f a;
      a.x = s_p[ml * 64 + kb * 4 + h];
      a.y = s_p[ml * 64 + (kb + 1) * 4 + h];
      const float* __restrict__ w0 = whb + (size_t)(j0 + kb) * CC + h * OF;
      const float* __restrict__ w1 = w0 + CC;
      v2f bf;
      bf.x = w0[ 0 + ml]; bf.y = w1[ 0 + ml];
      acc0 = wmma_f32_4(a, bf, acc0);
      bf.x = w0[16 + ml]; bf.y = w1[16 + ml];
      acc1 = wmma_f32_4(a, bf, acc1);
      bf.x = w0[32 + ml]; bf.y = w1[32 + ml];
      acc2 = wmma_f32_4(a, bf, acc2);
      bf.x = w0[48 + ml]; bf.y = w1[48 + ml];
      acc3 = wmma_f32_4(a, bf, acc3);
    }
    __syncthreads();
  }

  // write h_prime slab: C/D layout M = v + 8*half, N = lane&15
#pragma unroll
  for (int v = 0; v < 8; ++v) {
    const int r = i0 + v + half * 8;
    float* __restrict__ orow = out + (size_t)(b * NN + r) * CC + h * OF;
    orow[ 0 + ml] = acc0[v];
    orow[16 + ml] = acc1[v];
    orow[32 + ml] = acc2[v];
    orow[48 + ml] = acc3[v];
  }
}

// ---------------------------------------------------------------------------
extern "C" void kernel_launch(void* const* d_in, const int* in_sizes, int n_in,
                              void* d_out, int out_size, void* d_ws, size_t ws_size,
                              hipStream_t stream) {
  const float* x   = (const float*)d_in[0];   // (4,2048,128)
  const int*   adj = (const int*)  d_in[1];   // (4,2048,2048)
  const float* W   = (const float*)d_in[2];   // (128,256)
  const float* a   = (const float*)d_in[3];   // (128,4,1)

  float* out  = (float*)d_out;                         // (4,2048,256)
  float* attn = out + (size_t)BB * NN * CC;            // (4,2048,2048,4)

  char*  ws = (char*)d_ws;
  float* Wh = (float*)ws;                              //  8 MB
  float* ed = (float*)(ws + (size_t)BB * NN * CC * 4);         // 128 KB
  float* mS = ed + (size_t)BB * NN * HH;
  float* dS = mS + (size_t)BB * NN * HH;

  wh_gemm    <<<(BB * NN / 16) * (CC / 16) / 4, 128, 0, stream>>>(x, W, Wh);
  edst_kernel<<<BB * NN,                        128, 0, stream>>>(Wh, a, ed);
  row_stats  <<<BB * NN,                        256, 0, stream>>>(adj, ed, mS, dS);
  attn_agg   <<<BB * (NN / 16),                 128, 0, stream>>>(adj, ed, mS, dS, Wh, out, attn);
}